// LinkPredModel_74783970558466
// MI455X (gfx1250) — compile-verified
//
#include <hip/hip_runtime.h>

typedef __attribute__((ext_vector_type(2))) float v2f;
typedef __attribute__((ext_vector_type(8))) float v8f;

constexpr int D = 128;           // feature dim == hidden dim
constexpr int LDS_PITCH = 132;   // 128 + 4 pad to avoid LDS bank conflicts
constexpr float NEG_SLOPE = 0.2f;

// ---------------------------------------------------------------- degree
__global__ __launch_bounds__(256) void deg_kernel(const int* __restrict__ dst,
                                                  float* __restrict__ deg, int E) {
    int i = blockIdx.x * 256 + threadIdx.x;
    if (i < E) atomicAdd(&deg[dst[i]], 1.0f);
}

__global__ __launch_bounds__(256) void invdeg_kernel(float* __restrict__ deg, int N) {
    int i = blockIdx.x * 256 + threadIdx.x;
    if (i < N) deg[i] = 1.0f / fmaxf(deg[i], 1.0f);
}

// ------------------------------------------------------- edge scatter-add
// One wave covers 32 consecutive columns of one edge: coalesced 128B gather
// from h[src], fp32 atomics into agg[dst].
__global__ __launch_bounds__(256) void scatter_kernel(const float* __restrict__ h,
                                                      const int* __restrict__ src,
                                                      const int* __restrict__ dst,
                                                      float* __restrict__ agg,
                                                      long long total /* E*128 */) {
    long long gid = (long long)blockIdx.x * 256 + threadIdx.x;
    if (gid >= total) return;
    long long e = gid >> 7;          // edge id
    int c = (int)(gid & 127);        // column
    long long s = src[e];
    long long d = dst[e];
    atomicAdd(&agg[d * D + c], h[s * D + c]);
}

// ----------------------------------------------- fused SAGE layer via WMMA
// out[n,:] = (agg[n,:]*invdeg[n]) @ Wl + b + X[n,:] @ Wr   (+ optional leaky relu)
// Block: 256 threads = 8 waves. Block handles 16 rows x 128 cols; wave w owns
// the 16x16 tile at column 16*w. K=128 swept in steps of 4 with
// v_wmma_f32_16x16x4_f32 (fp32 end to end, matching the reference).
template <int RELU>
__global__ __launch_bounds__(256) void sage_gemm_kernel(const float* __restrict__ Agg,
                                                        const float* __restrict__ invdeg,
                                                        const float* __restrict__ X,
                                                        const float* __restrict__ Wl,
                                                        const float* __restrict__ bias,
                                                        const float* __restrict__ Wr,
                                                        float* __restrict__ Out,
                                                        int N) {
    __shared__ float lds1[16 * LDS_PITCH];   // scaled aggregate tile
    __shared__ float lds2[16 * LDS_PITCH];   // self-feature tile

    const int row0 = blockIdx.x * 16;
    const int tid = threadIdx.x;

    // ---- stage A tiles: each thread loads 8 consecutive floats of one row
    {
        const int r  = tid >> 4;          // tile row 0..15
        const int c0 = (tid & 15) * 8;    // column start
        const long long g = (long long)(row0 + r) * D + c0;
        const float s = invdeg[row0 + r];
        float4 p0 = *(const float4*)(Agg + g);
        float4 p1 = *(const float4*)(Agg + g + 4);
        float4 q0 = *(const float4*)(X + g);
        float4 q1 = *(const float4*)(X + g + 4);
        float4 sp0 = make_float4(p0.x * s, p0.y * s, p0.z * s, p0.w * s);
        float4 sp1 = make_float4(p1.x * s, p1.y * s, p1.z * s, p1.w * s);
        *(float4*)&lds1[r * LDS_PITCH + c0]     = sp0;
        *(float4*)&lds1[r * LDS_PITCH + c0 + 4] = sp1;
        *(float4*)&lds2[r * LDS_PITCH + c0]     = q0;
        *(float4*)&lds2[r * LDS_PITCH + c0 + 4] = q1;
    }
    __syncthreads();

    const int wv   = tid >> 5;        // wave 0..7 -> column tile
    const int lane = tid & 31;
    const int m    = lane & 15;       // A row / B,C,D column within tile
    const int half = lane >> 4;       // lane-half select (K split)
    const int j0   = wv * 16;

    v8f acc = {};
    const float* a1p = &lds1[m * LDS_PITCH + 2 * half];
    const float* a2p = &lds2[m * LDS_PITCH + 2 * half];

#pragma unroll
    for (int k0 = 0; k0 < D; k0 += 4) {
        const int kk = k0 + 2 * half;
        // A fragment (16x4 f32): a[v] = A[m][kk+v]
        v2f a1 = { a1p[k0], a1p[k0 + 1] };
        // B fragment (4x16 f32): b[v] = B[kk+v][j0+m]
        v2f wl = { Wl[(long long)kk * D + j0 + m], Wl[(long long)(kk + 1) * D + j0 + m] };
        acc = __builtin_amdgcn_wmma_f32_16x16x4_f32(false, a1, false, wl,
                                                    (short)0, acc, false, false);
        v2f a2 = { a2p[k0], a2p[k0 + 1] };
        v2f wr = { Wr[(long long)kk * D + j0 + m], Wr[(long long)(kk + 1) * D + j0 + m] };
        acc = __builtin_amdgcn_wmma_f32_16x16x4_f32(false, a2, false, wr,
                                                    (short)0, acc, false, false);
    }

    const float bv = bias[j0 + m];
#pragma unroll
    for (int rr = 0; rr < 8; rr++) {
        float v = acc[rr] + bv;
        if (RELU) v = (v >= 0.0f) ? v : NEG_SLOPE * v;
        Out[(long long)(row0 + rr + 8 * half) * D + j0 + m] = v;
    }
    (void)N;
}

// ----------------------------------------------------- link-pred dot pass
__global__ __launch_bounds__(256) void dot_kernel(const float* __restrict__ h,
                                                  const int* __restrict__ si,
                                                  const int* __restrict__ ti,
                                                  float* __restrict__ out, int L) {
    int wid  = (blockIdx.x * 256 + threadIdx.x) >> 5;
    int lane = threadIdx.x & 31;
    if (wid >= L) return;
    long long s = si[wid];
    long long t = ti[wid];
    const float* hs = h + s * D;
    const float* ht = h + t * D;
    float sum = 0.0f;
#pragma unroll
    for (int c = 0; c < 4; c++) sum += hs[lane + 32 * c] * ht[lane + 32 * c];
#pragma unroll
    for (int off = 16; off > 0; off >>= 1) sum += __shfl_down(sum, off, 32);
    if (lane == 0) out[wid] = sum;
}

// ---------------------------------------------------------------- launch
extern "C" void kernel_launch(void* const* d_in, const int* in_sizes, int n_in,
                              void* d_out, int out_size, void* d_ws, size_t ws_size,
                              hipStream_t stream) {
    const float* x   = (const float*)d_in[0];
    const int*   ei  = (const int*)d_in[1];   // edge_index (2,E)
    const int*   eli = (const int*)d_in[2];   // edge_label_index (2,L)
    const float* W1l = (const float*)d_in[3];
    const float* b1  = (const float*)d_in[4];
    const float* W1r = (const float*)d_in[5];
    const float* W2l = (const float*)d_in[6];
    const float* b2  = (const float*)d_in[7];
    const float* W2r = (const float*)d_in[8];
    float* out = (float*)d_out;

    const int       N = in_sizes[0] / D;
    const long long E = in_sizes[1] / 2;
    const int       L = in_sizes[2] / 2;
    const int* src = ei;
    const int* dst = ei + E;
    const int* ls  = eli;
    const int* lt  = eli + L;

    // workspace carve-out: deg | agg (reused both layers) | h1 | h2
    char*  ws  = (char*)d_ws;
    size_t off = 0;
    float* deg = (float*)(ws + off); off += (((size_t)N * 4) + 255) & ~(size_t)255;
    float* agg = (float*)(ws + off); off += (size_t)N * D * 4;
    float* h1  = (float*)(ws + off); off += (size_t)N * D * 4;
    float* h2  = (float*)(ws + off); off += (size_t)N * D * 4;
    (void)ws_size; (void)n_in; (void)out_size;

    const long long totalEC = E * D;
    const int degBlocks   = (int)((E + 255) / 256);
    const int nBlocks     = (N + 255) / 256;
    const int scatBlocks  = (int)((totalEC + 255) / 256);
    const int gemmBlocks  = N / 16;                 // N = 100000 is a multiple of 16
    const int dotBlocks   = (int)(((long long)L * 32 + 255) / 256);

    // degree + inverse degree
    hipMemsetAsync(deg, 0, (size_t)N * 4, stream);
    deg_kernel<<<degBlocks, 256, 0, stream>>>(dst, deg, (int)E);
    invdeg_kernel<<<nBlocks, 256, 0, stream>>>(deg, N);

    // layer 1
    hipMemsetAsync(agg, 0, (size_t)N * D * 4, stream);
    scatter_kernel<<<scatBlocks, 256, 0, stream>>>(x, src, dst, agg, totalEC);
    sage_gemm_kernel<1><<<gemmBlocks, 256, 0, stream>>>(agg, deg, x, W1l, b1, W1r, h1, N);

    // layer 2
    hipMemsetAsync(agg, 0, (size_t)N * D * 4, stream);
    scatter_kernel<<<scatBlocks, 256, 0, stream>>>(h1, src, dst, agg, totalEC);
    sage_gemm_kernel<0><<<gemmBlocks, 256, 0, stream>>>(agg, deg, h1, W2l, b2, W2r, h2, N);

    // link prediction dot
    dot_kernel<<<dotBlocks, 256, 0, stream>>>(h2, ls, lt, out, L);
}